// GAT_17239998726586
// MI455X (gfx1250) — compile-verified
//
#include <hip/hip_runtime.h>
#include <cstdint>

// ---------------------------------------------------------------------------
// GAT (4x GATConv + BN/ReLU + mean-pool) for MI455X / gfx1250, wave32.
// Dense transforms use V_WMMA_F32_16X16X4_F32 (fp32-exact matrix path);
// graph scatter phases are L2-resident atomics (192MB L2 >> working set).
//
// Round-1 refinement: W is staged into LDS in a K-pair-interleaved layout so
// each WMMA B-fragment (rows ka, ka+1 of one 16-wide column tile) is a single
// aligned ds_load_b64 into an even VGPR pair -> removes the v_dual_mov
// shuffle chains seen in the round-0 disassembly.
// ---------------------------------------------------------------------------

typedef __attribute__((ext_vector_type(2))) float v2f;
typedef __attribute__((ext_vector_type(8))) float v8f;

#define NN   50000            // nodes
#define NE   800000           // base edges
#define NET  (NE + NN)        // edges + self loops
#define WID  128              // heads*dout for the 3 wide layers
#define NG   500              // graphs
#define NEG_SLOPE 0.2f
#define BN_EPS 1e-5f
#define FLT_NEG_MAX -3.4028234663852886e38f

// ------------------------------ helpers ------------------------------------
__device__ __forceinline__ void atomicMaxF(float* addr, float val) {
  unsigned int* ai = reinterpret_cast<unsigned int*>(addr);
  unsigned int old = *ai;
  while (__uint_as_float(old) < val) {
    unsigned int assumed = old;
    old = atomicCAS(ai, assumed, __float_as_uint(val));
    if (old == assumed) break;
  }
}

__global__ void fill_f32(float* __restrict__ p, float v, int n) {
  int i = blockIdx.x * blockDim.x + threadIdx.x;
  if (i < n) p[i] = v;
}

// ------------------- fp32 GEMM via WMMA: H = X[rows,128] * W[128,128] -------
// Wave computes a 16x128 stripe of H: 8 N-tiles, 32 K-steps of 16x16x4 WMMA.
// LDS layout: Wl[((k>>1)*128 + col)*2 + (k&1)] = W[k][col]  (K-pair interleave)
// so B fragment for (ka, col) is the 64-bit word at pair index (ka>>1)*128+col.
__global__ __launch_bounds__(256) void linear128_wmma(
    const float* __restrict__ X, const float* __restrict__ W,
    float* __restrict__ H, int n_rows)
{
  __shared__ float Wl[128 * 128];
  for (int j = threadIdx.x; j < 128 * 128; j += 256) {
    int t = j >> 1;
    int c = t & 127;                       // column
    int k = ((t >> 7) << 1) | (j & 1);     // K row
    Wl[j] = W[k * 128 + c];
  }
  __syncthreads();

  const int lane = threadIdx.x & 31;
  const int wave = threadIdx.x >> 5;
  const int row0 = (blockIdx.x * 8 + wave) * 16;
  if (row0 >= n_rows) return;

  const int mrow  = lane & 15;   // A: lanes 0-15 rows, repeated for 16-31
  const int khalf = lane >> 4;   // lanes 16-31 hold K+2..K+3 (ISA 7.12.2)
  const int ncol  = lane & 15;   // B/C: column within 16-wide tile

  v8f acc[8] = {};

  const float* __restrict__ arow = X + (size_t)(row0 + mrow) * 128;
  const v2f* __restrict__ Wl2 = (const v2f*)Wl;   // pair-indexed view
  for (int k = 0; k < 128; k += 4) {
    const int ka = k + khalf * 2;
    v2f a;
    a.x = arow[ka + 0];
    a.y = arow[ka + 1];
    const int pbase = (ka >> 1) * 128 + ncol;     // pair index of (ka, ncol)
#pragma unroll
    for (int t = 0; t < 8; ++t) {
      v2f b = Wl2[pbase + t * 16];                // single ds_load_b64
      acc[t] = __builtin_amdgcn_wmma_f32_16x16x4_f32(
          false, a, false, b, (short)0, acc[t], false, false);
    }
  }
  // C/D layout: VGPR r -> row M = r + 8*khalf
#pragma unroll
  for (int t = 0; t < 8; ++t)
#pragma unroll
    for (int r = 0; r < 8; ++r)
      H[(size_t)(row0 + r + khalf * 8) * 128 + t * 16 + ncol] = acc[t][r];
}

// Final layer linear: dout_total = 1
__global__ void linear_n1(const float* __restrict__ X, const float* __restrict__ W,
                          float* __restrict__ H, int n_rows) {
  int n = blockIdx.x * blockDim.x + threadIdx.x;
  if (n >= n_rows) return;
  const float* xr = X + (size_t)n * 128;
  float acc = 0.f;
#pragma unroll 4
  for (int k = 0; k < 128; ++k) acc += xr[k] * W[k];
  H[n] = acc;
}

// -------------------- per-node attention logits -----------------------------
// a_s[n,h] = sum_c h[n,h,c]*a_src[h,c] ; a_d likewise
__global__ void node_scores(const float* __restrict__ h,
                            const float* __restrict__ a_src,
                            const float* __restrict__ a_dst,
                            float* __restrict__ as_, float* __restrict__ ad_,
                            int n_nodes, int heads, int dout) {
  int idx = blockIdx.x * blockDim.x + threadIdx.x;
  if (idx >= n_nodes * heads) return;
  int n = idx / heads, hd = idx % heads;
  const float* hp = h + (size_t)n * heads * dout + (size_t)hd * dout;
  float s1 = 0.f, s2 = 0.f;
  for (int c = 0; c < dout; ++c) {
    float v = hp[c];
    s1 += v * a_src[hd * dout + c];
    s2 += v * a_dst[hd * dout + c];
  }
  as_[idx] = s1;
  ad_[idx] = s2;
}

// -------------------- edge passes (segment softmax) -------------------------
__global__ void edge_scores(const float* __restrict__ as_, const float* __restrict__ ad_,
                            const int* __restrict__ ei, float* __restrict__ ev,
                            float* __restrict__ mmax,
                            int n_base, int n_total, int heads) {
  int idx = blockIdx.x * blockDim.x + threadIdx.x;
  if (idx >= n_total * heads) return;
  int e = idx / heads, hd = idx % heads;
  int src, dst;
  if (e < n_base) { src = ei[e]; dst = ei[n_base + e]; }
  else            { src = dst = e - n_base; }
  float v = as_[src * heads + hd] + ad_[dst * heads + hd];
  v = v >= 0.f ? v : NEG_SLOPE * v;       // leaky relu
  ev[idx] = v;
  atomicMaxF(&mmax[dst * heads + hd], v); // segment max
}

__global__ void edge_exp(const float* __restrict__ ev, const float* __restrict__ mmax,
                         const int* __restrict__ ei, float* __restrict__ pv,
                         float* __restrict__ ssum,
                         int n_base, int n_total, int heads) {
  int idx = blockIdx.x * blockDim.x + threadIdx.x;
  if (idx >= n_total * heads) return;
  int e = idx / heads, hd = idx % heads;
  int dst = (e < n_base) ? ei[n_base + e] : (e - n_base);
  float p = expf(ev[idx] - mmax[dst * heads + hd]);
  pv[idx] = p;
  atomicAdd(&ssum[dst * heads + hd], p);  // segment sum
}

// -------------------- weighted scatter (wave per edge, width=128) -----------
// heads=2, dout=64: lanes 0-15 -> head0 channels, 16-31 -> head1. float4/lane.
__global__ __launch_bounds__(256) void edge_aggregate128(
    const float* __restrict__ pv, const float* __restrict__ ssum,
    const float* __restrict__ h, const int* __restrict__ ei,
    float* __restrict__ out, int n_base, int n_total)
{
  int e = blockIdx.x * 8 + (threadIdx.x >> 5);
  if (e >= n_total) return;
  int lane = threadIdx.x & 31;
  int src, dst;
  if (e < n_base) { src = ei[e]; dst = ei[n_base + e]; }
  else            { src = dst = e - n_base; }
  int head = lane >> 4;
  float alpha = pv[e * 2 + head] / (ssum[dst * 2 + head] + 1e-16f);
  int c = lane * 4;
  const float4 hv = *(const float4*)(h + (size_t)src * 128 + c);
  float* o = out + (size_t)dst * 128 + c;
  atomicAdd(o + 0, alpha * hv.x);
  atomicAdd(o + 1, alpha * hv.y);
  atomicAdd(o + 2, alpha * hv.z);
  atomicAdd(o + 3, alpha * hv.w);
}

// Final layer scatter: width = 1, heads = 1
__global__ void edge_aggregate1(const float* __restrict__ pv, const float* __restrict__ ssum,
                                const float* __restrict__ h, const int* __restrict__ ei,
                                float* __restrict__ out, int n_base, int n_total) {
  int e = blockIdx.x * blockDim.x + threadIdx.x;
  if (e >= n_total) return;
  int src, dst;
  if (e < n_base) { src = ei[e]; dst = ei[n_base + e]; }
  else            { src = dst = e - n_base; }
  float alpha = pv[e] / (ssum[dst] + 1e-16f);
  atomicAdd(&out[dst], alpha * h[src]);
}

// -------------------- batchnorm (bias fused in) -----------------------------
// stats[0..127] = sum, stats[128..255] = sumsq. One thread per channel per
// block, register accumulation over strided nodes, one atomic per thread.
__global__ __launch_bounds__(128) void bn_stats(const float* __restrict__ out,
                                                const float* __restrict__ bias,
                                                float* __restrict__ stats, int n_nodes) {
  int c = threadIdx.x;
  float b = bias[c];
  float sum = 0.f, sq = 0.f;
  for (int n = blockIdx.x; n < n_nodes; n += gridDim.x) {
    float v = out[(size_t)n * 128 + c] + b;
    sum += v; sq += v * v;
  }
  atomicAdd(&stats[c], sum);
  atomicAdd(&stats[128 + c], sq);
}

__global__ void bn_apply(const float* __restrict__ out, const float* __restrict__ bias,
                         const float* __restrict__ stats,
                         const float* __restrict__ gamma, const float* __restrict__ beta,
                         float* __restrict__ hnext, int n_nodes) {
  int idx = blockIdx.x * blockDim.x + threadIdx.x;
  if (idx >= n_nodes * 128) return;
  int c = idx & 127;
  float inv_n = 1.f / (float)n_nodes;
  float mu  = stats[c] * inv_n;
  float var = stats[128 + c] * inv_n - mu * mu;   // biased var, torch BN
  float v = out[idx] + bias[c];
  float y = gamma[c] * (v - mu) * rsqrtf(var + BN_EPS) + beta[c];
  hnext[idx] = fmaxf(y, 0.f);                     // relu
}

// -------------------- scatter-mean pooling ----------------------------------
__global__ void pool_accum(const float* __restrict__ hfin, const float* __restrict__ bias,
                           const int* __restrict__ batch, float* __restrict__ pool,
                           int n_nodes) {
  int n = blockIdx.x * blockDim.x + threadIdx.x;
  if (n >= n_nodes) return;
  int g = batch[n];
  atomicAdd(&pool[g], hfin[n] + bias[0]);   // final conv bias applied here
  atomicAdd(&pool[NG + g], 1.f);
}

__global__ void pool_final(const float* __restrict__ pool, float* __restrict__ out, int ng) {
  int g = blockIdx.x * blockDim.x + threadIdx.x;
  if (g >= ng) return;
  out[g] = pool[g] / fmaxf(pool[NG + g], 1.f);
}

// ---------------------------------------------------------------------------
extern "C" void kernel_launch(void* const* d_in, const int* in_sizes, int n_in,
                              void* d_out, int out_size, void* d_ws, size_t ws_size,
                              hipStream_t stream) {
  (void)in_sizes; (void)n_in; (void)out_size; (void)ws_size;

  // setup_inputs() dict order: x, edge_index, batch_ids, then params leaves
  // (convs[0..3]: W, a_src, a_dst, b; bns[0..2]: gamma, beta)
  const float* x   = (const float*)d_in[0];
  const int*   ei  = (const int*)d_in[1];
  const int*   bid = (const int*)d_in[2];
  const float* Wp[4] = {(const float*)d_in[3],  (const float*)d_in[7],
                        (const float*)d_in[11], (const float*)d_in[15]};
  const float* As[4] = {(const float*)d_in[4],  (const float*)d_in[8],
                        (const float*)d_in[12], (const float*)d_in[16]};
  const float* Ad[4] = {(const float*)d_in[5],  (const float*)d_in[9],
                        (const float*)d_in[13], (const float*)d_in[17]};
  const float* Bp[4] = {(const float*)d_in[6],  (const float*)d_in[10],
                        (const float*)d_in[14], (const float*)d_in[18]};
  const float* Gm[3] = {(const float*)d_in[19], (const float*)d_in[21], (const float*)d_in[23]};
  const float* Bt[3] = {(const float*)d_in[20], (const float*)d_in[22], (const float*)d_in[24]};

  // workspace layout (floats)
  float* w = (float*)d_ws;
  float* featA = w;  w += (size_t)NN * WID;   // h_next (layer input after BN)
  float* featB = w;  w += (size_t)NN * WID;   // h = x@W
  float* featC = w;  w += (size_t)NN * WID;   // aggregation accumulator
  float* a_s   = w;  w += (size_t)NN * 2;
  float* a_d   = w;  w += (size_t)NN * 2;
  float* mmax  = w;  w += (size_t)NN * 2;
  float* ssum  = w;  w += (size_t)NN * 2;
  float* ev    = w;  w += (size_t)NET * 2;
  float* pv    = w;  w += (size_t)NET * 2;
  float* stats = w;  w += 256;
  float* pool  = w;  w += 2 * NG;

  const int gemmGrid = ((NN / 16) + 7) / 8;            // 16-row tiles, 8 waves/block

  // --------- three wide GATConv layers (heads=2, dout=64) ---------
  for (int i = 0; i < 3; ++i) {
    const float* cur = (i == 0) ? x : featA;

    linear128_wmma<<<gemmGrid, 256, 0, stream>>>(cur, Wp[i], featB, NN);
    node_scores<<<(NN * 2 + 255) / 256, 256, 0, stream>>>(featB, As[i], Ad[i],
                                                          a_s, a_d, NN, 2, 64);
    fill_f32<<<(NN * 2 + 255) / 256, 256, 0, stream>>>(mmax, FLT_NEG_MAX, NN * 2);
    fill_f32<<<(NN * 2 + 255) / 256, 256, 0, stream>>>(ssum, 0.f, NN * 2);
    fill_f32<<<(NN * WID + 255) / 256, 256, 0, stream>>>(featC, 0.f, NN * WID);

    edge_scores<<<(NET * 2 + 255) / 256, 256, 0, stream>>>(a_s, a_d, ei, ev, mmax,
                                                           NE, NET, 2);
    edge_exp<<<(NET * 2 + 255) / 256, 256, 0, stream>>>(ev, mmax, ei, pv, ssum,
                                                        NE, NET, 2);
    edge_aggregate128<<<(NET + 7) / 8, 256, 0, stream>>>(pv, ssum, featB, ei, featC,
                                                         NE, NET);

    fill_f32<<<1, 256, 0, stream>>>(stats, 0.f, 256);
    bn_stats<<<512, 128, 0, stream>>>(featC, Bp[i], stats, NN);
    bn_apply<<<(NN * WID + 255) / 256, 256, 0, stream>>>(featC, Bp[i], stats,
                                                         Gm[i], Bt[i], featA, NN);
  }

  // --------- final GATConv (heads=1, dout=1) + mean pool ---------
  float* hfin = featB;
  float* ofin = featC;
  linear_n1<<<(NN + 255) / 256, 256, 0, stream>>>(featA, Wp[3], hfin, NN);
  node_scores<<<(NN + 255) / 256, 256, 0, stream>>>(hfin, As[3], Ad[3],
                                                    a_s, a_d, NN, 1, 1);
  fill_f32<<<(NN + 255) / 256, 256, 0, stream>>>(mmax, FLT_NEG_MAX, NN);
  fill_f32<<<(NN + 255) / 256, 256, 0, stream>>>(ssum, 0.f, NN);
  fill_f32<<<(NN + 255) / 256, 256, 0, stream>>>(ofin, 0.f, NN);

  edge_scores<<<(NET + 255) / 256, 256, 0, stream>>>(a_s, a_d, ei, ev, mmax,
                                                     NE, NET, 1);
  edge_exp<<<(NET + 255) / 256, 256, 0, stream>>>(ev, mmax, ei, pv, ssum,
                                                  NE, NET, 1);
  edge_aggregate1<<<(NET + 255) / 256, 256, 0, stream>>>(pv, ssum, hfin, ei, ofin,
                                                         NE, NET);

  fill_f32<<<(2 * NG + 255) / 256, 256, 0, stream>>>(pool, 0.f, 2 * NG);
  pool_accum<<<(NN + 255) / 256, 256, 0, stream>>>(ofin, Bp[3], bid, pool, NN);
  pool_final<<<(NG + 255) / 256, 256, 0, stream>>>(pool, (float*)d_out, NG);
}